// FWPBlock_9405978378327
// MI455X (gfx1250) — compile-verified
//
#include <hip/hip_runtime.h>
#include <hip/hip_bf16.h>

typedef __attribute__((ext_vector_type(2))) float v2f;
typedef __attribute__((ext_vector_type(4))) float v4f;
typedef __attribute__((ext_vector_type(8))) float v8f;

#define NB 8
#define NT 1024
#define NF 128
#define NH 64
#define TC 32              // chunk length for the hierarchical scan
#define NC (NT / TC)       // 32 chunks per batch
#define EPS_LN 1e-5f
#define EPS_SN 1e-5f

// -------------------------------------------------------------------------
// Kernel 1: LayerNorm + 4 projections (K relu+sumnorm, Q relu+sumnorm, V,
// residual xn@Ws^T+bs) using V_WMMA_F32_16X16X4_F32.
// One WG = 256 threads = 8 waves handles 64 rows of (B*T).
// -------------------------------------------------------------------------
__global__ __launch_bounds__(256) void k_proj(
    const float* __restrict__ x, const float* __restrict__ gamma,
    const float* __restrict__ beta,
    const float* __restrict__ Wk, const float* __restrict__ Wq,
    const float* __restrict__ Wv, const float* __restrict__ Ws,
    const float* __restrict__ bs,
    float* __restrict__ Kb, float* __restrict__ Qb,
    float* __restrict__ Vb, float* __restrict__ Rb)
{
    __shared__ float xn[64 * NF];              // 32 KB normalized tile
    const int tid = threadIdx.x;
    const int rowBase = blockIdx.x * 64;

    // ---- LayerNorm: 4 lanes per row, 32 features each ----
    {
        const int r = tid >> 2;
        const int seg = tid & 3;
        const float* xp = x + (size_t)(rowBase + r) * NF + seg * 32;
        float vals[32];
        float sum = 0.f, sq = 0.f;
        #pragma unroll
        for (int i = 0; i < 32; ++i) { float v = xp[i]; vals[i] = v; sum += v; sq += v * v; }
        sum += __shfl_xor(sum, 1, 32); sq += __shfl_xor(sq, 1, 32);
        sum += __shfl_xor(sum, 2, 32); sq += __shfl_xor(sq, 2, 32);
        const float mean = sum * (1.0f / NF);
        const float var  = sq  * (1.0f / NF) - mean * mean;
        const float rstd = rsqrtf(var + EPS_LN);
        #pragma unroll
        for (int i = 0; i < 32; ++i) {
            const int f = seg * 32 + i;
            xn[r * NF + f] = (vals[i] - mean) * rstd * gamma[f] + beta[f];
        }
    }
    __syncthreads();

    // ---- WMMA GEMMs: combo = (matrix, mtile); each wave does 2 combos ----
    const int wave = tid >> 5;
    const int lane = tid & 31;
    const int row  = lane & 15;    // M (A) or N (B/D) within tile
    const int hi   = lane >> 4;    // K-half for A/B, +8 rows for D
    const int kofs = hi * 2;

    for (int cb = wave; cb < 16; cb += 8) {
        const int mat   = cb >> 2;     // 0=K 1=Q 2=V 3=Res
        const int mtile = cb & 3;
        const float* W = (mat == 0) ? Wk : (mat == 1) ? Wq : (mat == 2) ? Wv : Ws;

        v8f acc[4] = {};               // 4 N-tiles of 16
        const float* aBase = &xn[(mtile * 16 + row) * NF];
        #pragma unroll 4
        for (int k0 = 0; k0 < NF; k0 += 4) {
            v2f a; a.x = aBase[k0 + kofs]; a.y = aBase[k0 + kofs + 1];
            #pragma unroll
            for (int nt = 0; nt < 4; ++nt) {
                const float* bp = W + (nt * 16 + row) * NF + k0 + kofs;
                v2f bfr; bfr.x = bp[0]; bfr.y = bp[1];
                acc[nt] = __builtin_amdgcn_wmma_f32_16x16x4_f32(
                    false, a, false, bfr, (short)0, acc[nt], false, false);
            }
        }

        float* outp;
        if (mat <= 1) {
            // relu then sum-normalize each row of 64 (N striped over lanes + ntiles)
            #pragma unroll
            for (int nt = 0; nt < 4; ++nt)
                #pragma unroll
                for (int rr = 0; rr < 8; ++rr) acc[nt][rr] = fmaxf(acc[nt][rr], 0.f);
            #pragma unroll
            for (int rr = 0; rr < 8; ++rr) {
                float s = acc[0][rr] + acc[1][rr] + acc[2][rr] + acc[3][rr];
                s += __shfl_xor(s, 1, 32); s += __shfl_xor(s, 2, 32);
                s += __shfl_xor(s, 4, 32); s += __shfl_xor(s, 8, 32);
                const float scale = 1.0f / (EPS_SN + s);
                #pragma unroll
                for (int nt = 0; nt < 4; ++nt) acc[nt][rr] *= scale;
            }
            outp = (mat == 0) ? Kb : Qb;
        } else if (mat == 3) {
            #pragma unroll
            for (int nt = 0; nt < 4; ++nt) {
                const float bv = bs[nt * 16 + row];
                #pragma unroll
                for (int rr = 0; rr < 8; ++rr) acc[nt][rr] += bv;
            }
            outp = Rb;
        } else {
            outp = Vb;
        }

        #pragma unroll
        for (int nt = 0; nt < 4; ++nt)
            #pragma unroll
            for (int rr = 0; rr < 8; ++rr) {
                const int M = rowBase + mtile * 16 + rr + 8 * hi;
                outp[(size_t)M * NH + nt * 16 + row] = acc[nt][rr];
            }
    }
}

// -------------------------------------------------------------------------
// Kernel 2: per-chunk outer-product sums  chunkSum = V_chunk^T @ K_chunk
// (64x64 GEMM with K=TC). One WG of 128 threads per (b,chunk), one wave
// per 16-row M-tile.
// -------------------------------------------------------------------------
__global__ __launch_bounds__(128) void k_chunksum(
    const float* __restrict__ Kb, const float* __restrict__ Vb,
    float* __restrict__ csum)
{
    const int bc = blockIdx.x;                  // b*NC + c
    const int b = bc / NC, c = bc % NC;
    const int t0 = c * TC;
    const int wave = threadIdx.x >> 5, lane = threadIdx.x & 31;
    const int row = lane & 15, hi = lane >> 4, kofs = hi * 2;
    const int mtile = wave;

    const float* Vc = Vb + (size_t)(b * NT + t0) * NH;   // [TC][H]
    const float* Kc = Kb + (size_t)(b * NT + t0) * NH;

    v8f acc[4] = {};
    #pragma unroll 2
    for (int k0 = 0; k0 < TC; k0 += 4) {
        const int kk = k0 + kofs;
        v2f a;
        a.x = Vc[(size_t)kk * NH + mtile * 16 + row];
        a.y = Vc[(size_t)(kk + 1) * NH + mtile * 16 + row];
        #pragma unroll
        for (int nt = 0; nt < 4; ++nt) {
            v2f bfr;
            bfr.x = Kc[(size_t)kk * NH + nt * 16 + row];
            bfr.y = Kc[(size_t)(kk + 1) * NH + nt * 16 + row];
            acc[nt] = __builtin_amdgcn_wmma_f32_16x16x4_f32(
                false, a, false, bfr, (short)0, acc[nt], false, false);
        }
    }

    float* outp = csum + (size_t)bc * (NH * NH);
    #pragma unroll
    for (int nt = 0; nt < 4; ++nt)
        #pragma unroll
        for (int rr = 0; rr < 8; ++rr)
            outp[(mtile * 16 + rr + 8 * hi) * NH + nt * 16 + row] = acc[nt][rr];
}

// -------------------------------------------------------------------------
// Kernel 3: exclusive scan over chunks, seeded with `state`.
// One thread per (b, i, j) element; 32 sequential chunk steps.
// -------------------------------------------------------------------------
__global__ __launch_bounds__(256) void k_scan(
    const float* __restrict__ state, const float* __restrict__ csum,
    float* __restrict__ offs)
{
    const int gid = blockIdx.x * 256 + threadIdx.x;     // B*H*H threads
    const int b = gid / (NH * NH);
    const int ij = gid % (NH * NH);
    float acc = state[(size_t)b * (NH * NH) + ij];
    for (int c = 0; c < NC; ++c) {
        const size_t idx = ((size_t)(b * NC + c)) * (NH * NH) + ij;
        offs[idx] = acc;
        acc += csum[idx];
    }
}

// -------------------------------------------------------------------------
// Kernel 4: within-chunk running sum; stream S to d_out with NT store hint
// (134 MB write-once stream must not thrash L2), K/Q/V served from LDS so
// the VMEM pipe is dedicated to the S store stream.
// One WG = 256 threads per (b,chunk); thread owns row i = tid/4, 16 cols.
// -------------------------------------------------------------------------
__global__ __launch_bounds__(256) void k_main(
    const float* __restrict__ Kb, const float* __restrict__ Qb,
    const float* __restrict__ Vb, const float* __restrict__ offs,
    float* __restrict__ Sout, float* __restrict__ ylin)
{
    __shared__ float Kc[TC * NH];    // 8 KB
    __shared__ float Qc[TC * NH];    // 8 KB
    __shared__ float Vc[TC * NH];    // 8 KB
    const int bc = blockIdx.x;
    const int b = bc / NC, c = bc % NC;
    const int t0 = c * TC;
    const int tid = threadIdx.x;

    // ---- cooperative stage of the whole chunk's K/Q/V into LDS ----
    {
        const size_t gbase = (size_t)(b * NT + t0) * NH;   // 16B-aligned
        const v4f* gK = (const v4f*)(Kb + gbase);
        const v4f* gQ = (const v4f*)(Qb + gbase);
        const v4f* gV = (const v4f*)(Vb + gbase);
        v4f* sK = (v4f*)Kc;
        v4f* sQ = (v4f*)Qc;
        v4f* sV = (v4f*)Vc;
        #pragma unroll
        for (int u = 0; u < 2; ++u) {          // 512 v4f per array
            const int idx = u * 256 + tid;
            sK[idx] = gK[idx];
            sQ[idx] = gQ[idx];
            sV[idx] = gV[idx];
        }
    }
    __syncthreads();

    const int i = tid >> 2;
    const int jb = (tid & 3) * 16;

    float S[16];
    {
        const v4f* op = (const v4f*)(offs + (size_t)bc * (NH * NH) + i * NH + jb);
        #pragma unroll
        for (int m = 0; m < 4; ++m) {
            v4f o = op[m];
            S[4 * m + 0] = o.x; S[4 * m + 1] = o.y; S[4 * m + 2] = o.z; S[4 * m + 3] = o.w;
        }
    }

    for (int lt = 0; lt < TC; ++lt) {
        const size_t t = (size_t)(b * NT + t0 + lt);
        const v4f* kp = (const v4f*)(Kc + lt * NH + jb);
        const v4f* qp = (const v4f*)(Qc + lt * NH + jb);
        const float v = Vc[lt * NH + i];
        v4f* sp = (v4f*)(Sout + t * (NH * NH) + i * NH + jb);

        float p = 0.f;
        #pragma unroll
        for (int m = 0; m < 4; ++m) {
            const v4f kk = kp[m];
            const v4f qq = qp[m];
            S[4 * m + 0] = fmaf(v, kk.x, S[4 * m + 0]);
            S[4 * m + 1] = fmaf(v, kk.y, S[4 * m + 1]);
            S[4 * m + 2] = fmaf(v, kk.z, S[4 * m + 2]);
            S[4 * m + 3] = fmaf(v, kk.w, S[4 * m + 3]);
            p = fmaf(S[4 * m + 0], qq.x, p);
            p = fmaf(S[4 * m + 1], qq.y, p);
            p = fmaf(S[4 * m + 2], qq.z, p);
            p = fmaf(S[4 * m + 3], qq.w, p);
            v4f s4;
            s4.x = S[4 * m + 0]; s4.y = S[4 * m + 1];
            s4.z = S[4 * m + 2]; s4.w = S[4 * m + 3];
            __builtin_nontemporal_store(s4, sp + m);   // stream S, keep L2 clean
        }
        p += __shfl_xor(p, 1, 32);
        p += __shfl_xor(p, 2, 32);
        if ((tid & 3) == 0) ylin[t * NH + i] = p;
    }
}

// -------------------------------------------------------------------------
// Kernel 5: FFN (two 64x64 layers, bias+ReLU) + residual, all WMMA.
// Layer-1 D tile is staged through per-wave LDS to become layer-2's A tile.
// One WG = 256 threads = 8 waves x 16 rows = 128 rows per WG.
// -------------------------------------------------------------------------
__global__ __launch_bounds__(256) void k_ffn(
    const float* __restrict__ ylin, const float* __restrict__ W1,
    const float* __restrict__ b1, const float* __restrict__ W2,
    const float* __restrict__ b2, const float* __restrict__ Rb,
    float* __restrict__ yout)
{
    __shared__ float y1s[8 * 16 * NH];   // 32 KB: per-wave 16x64 tile
    const int wave = threadIdx.x >> 5, lane = threadIdx.x & 31;
    const int row = lane & 15, hi = lane >> 4, kofs = hi * 2;
    const int mbase = blockIdx.x * 128 + wave * 16;
    float* my1 = &y1s[wave * 16 * NH];

    // ---- layer 1: y1 = relu(ylin @ W1^T + b1) ----
    v8f acc[4] = {};
    #pragma unroll 4
    for (int k0 = 0; k0 < NH; k0 += 4) {
        v2f a;
        a.x = ylin[(size_t)(mbase + row) * NH + k0 + kofs];
        a.y = ylin[(size_t)(mbase + row) * NH + k0 + kofs + 1];
        #pragma unroll
        for (int nt = 0; nt < 4; ++nt) {
            const float* bp = W1 + (nt * 16 + row) * NH + k0 + kofs;
            v2f bfr; bfr.x = bp[0]; bfr.y = bp[1];
            acc[nt] = __builtin_amdgcn_wmma_f32_16x16x4_f32(
                false, a, false, bfr, (short)0, acc[nt], false, false);
        }
    }
    #pragma unroll
    for (int nt = 0; nt < 4; ++nt) {
        const float bv = b1[nt * 16 + row];
        #pragma unroll
        for (int rr = 0; rr < 8; ++rr)
            my1[(rr + 8 * hi) * NH + nt * 16 + row] = fmaxf(acc[nt][rr] + bv, 0.f);
    }

    // ---- layer 2: y2 = relu(y1 @ W2^T + b2) + res ----
    v8f acc2[4] = {};
    #pragma unroll 4
    for (int k0 = 0; k0 < NH; k0 += 4) {
        v2f a;
        a.x = my1[row * NH + k0 + kofs];
        a.y = my1[row * NH + k0 + kofs + 1];
        #pragma unroll
        for (int nt = 0; nt < 4; ++nt) {
            const float* bp = W2 + (nt * 16 + row) * NH + k0 + kofs;
            v2f bfr; bfr.x = bp[0]; bfr.y = bp[1];
            acc2[nt] = __builtin_amdgcn_wmma_f32_16x16x4_f32(
                false, a, false, bfr, (short)0, acc2[nt], false, false);
        }
    }
    #pragma unroll
    for (int nt = 0; nt < 4; ++nt) {
        const float bv = b2[nt * 16 + row];
        #pragma unroll
        for (int rr = 0; rr < 8; ++rr) {
            const size_t M = mbase + rr + 8 * hi;
            const float res = Rb[M * NH + nt * 16 + row];
            yout[M * NH + nt * 16 + row] = fmaxf(acc2[nt][rr] + bv, 0.f) + res;
        }
    }
}

// -------------------------------------------------------------------------
extern "C" void kernel_launch(void* const* d_in, const int* in_sizes, int n_in,
                              void* d_out, int out_size, void* d_ws, size_t ws_size,
                              hipStream_t stream) {
    const float* x     = (const float*)d_in[0];
    const float* state = (const float*)d_in[1];
    const float* Wk    = (const float*)d_in[2];
    const float* Wq    = (const float*)d_in[3];
    const float* Wv    = (const float*)d_in[4];
    const float* gamma = (const float*)d_in[5];
    const float* beta  = (const float*)d_in[6];
    const float* W1    = (const float*)d_in[7];
    const float* b1    = (const float*)d_in[8];
    const float* W2    = (const float*)d_in[9];
    const float* b2    = (const float*)d_in[10];
    const float* Ws    = (const float*)d_in[11];
    const float* bs    = (const float*)d_in[12];

    float* yout = (float*)d_out;                       // [B,T,H]
    float* Sout = yout + (size_t)NB * NT * NH;         // [B,T,H,H]

    const size_t NR = (size_t)NB * NT;                 // 8192 rows
    float* ws   = (float*)d_ws;
    float* Kb   = ws;
    float* Qb   = Kb + NR * NH;
    float* Vb   = Qb + NR * NH;
    float* Rb   = Vb + NR * NH;
    float* ylin = Rb + NR * NH;
    float* csum = ylin + NR * NH;                      // [B*NC, H*H]
    float* offs = csum + (size_t)NB * NC * NH * NH;    // [B*NC, H*H]

    k_proj<<<dim3(NR / 64), dim3(256), 0, stream>>>(
        x, gamma, beta, Wk, Wq, Wv, Ws, bs, Kb, Qb, Vb, Rb);

    k_chunksum<<<dim3(NB * NC), dim3(128), 0, stream>>>(Kb, Vb, csum);

    k_scan<<<dim3((NB * NH * NH) / 256), dim3(256), 0, stream>>>(state, csum, offs);

    k_main<<<dim3(NB * NC), dim3(256), 0, stream>>>(Kb, Qb, Vb, offs, Sout, ylin);

    k_ffn<<<dim3(NR / 128), dim3(256), 0, stream>>>(ylin, W1, b1, W2, b2, Rb, yout);
}